// EncoderGCN_18940805775457
// MI455X (gfx1250) — compile-verified
//
#include <hip/hip_runtime.h>

typedef float v2f __attribute__((ext_vector_type(2)));
typedef float v8f __attribute__((ext_vector_type(8)));

// ---------------------------------------------------------------- utilities

__global__ void zero_kernel(float* __restrict__ p, long long n) {
  long long i = (long long)blockIdx.x * blockDim.x + threadIdx.x;
  if (i < n) p[i] = 0.f;
}

// Pad x (N x 11, row-major) into X16 (N x 16, zero-padded columns 11..15).
__global__ void pad_x_kernel(const float* __restrict__ x, float* __restrict__ x16,
                             int n) {
  long long t = (long long)blockIdx.x * blockDim.x + threadIdx.x;
  int i = (int)(t >> 4);
  int f = (int)(t & 15);
  if (i >= n) return;
  x16[(size_t)i * 16 + f] = (f < 11) ? x[(size_t)i * 11 + f] : 0.f;
}

// Pack W (fin x fout, row-major) into WMMA B-operand lane order, zero-padded.
// Chunk c covers K = 4c..4c+3.  Lane holds column ncol = lane&15 and the
// K-pair starting at 4c + 2*(lane>>4):  Bp[c*64 + lane*2 + j] = W[k][ncol],
// k = 4c + 2*(lane>>4) + j.  Each lane then loads its pair with one b64 load.
__global__ void pack_w_kernel(const float* __restrict__ W, float* __restrict__ Bp,
                              int fin, int fout, int chunks) {
  int t = blockIdx.x * blockDim.x + threadIdx.x;
  if (t >= chunks * 64) return;
  int c    = t >> 6;
  int r    = t & 63;
  int lane = r >> 1;
  int j    = r & 1;
  int ncol = lane & 15;
  int k    = c * 4 + ((lane >> 4) << 1) + j;
  Bp[t] = (k < fin && ncol < fout) ? W[k * fout + ncol] : 0.f;
}

__global__ void degree_kernel(const long long* __restrict__ dst,
                              float* __restrict__ deg, long long nedges) {
  long long e = (long long)blockIdx.x * blockDim.x + threadIdx.x;
  if (e >= nedges) return;
  atomicAdd(deg + (int)dst[e], 1.0f);
}

__global__ void dinv_kernel(float* __restrict__ degdinv, int n) {
  int i = blockIdx.x * blockDim.x + threadIdx.x;
  if (i >= n) return;
  degdinv[i] = rsqrtf(degdinv[i] + 1.0f);  // self-loop
}

// ------------------------------------------------- WMMA dense transform
// h[N x 16] = in[N x 16] @ B   (B pre-packed in lane order, zero-padded).
// One wave per 16-row tile: clause of 2*CHUNKS global_load_b64, CHUNKS
// back-to-back v_wmma_f32_16x16x4_f32, then 8 straight global_store_b32
// (full-tile fast path is wave-uniform, so EXEC stays all-1s throughout).
template <int CHUNKS>
__global__ __launch_bounds__(256)
void matmul_wmma_kernel(const float* __restrict__ in,
                        const float* __restrict__ Bp,
                        float* __restrict__ h, int n) {
  int wave = blockIdx.x * (blockDim.x >> 5) + (threadIdx.x >> 5);
  int lane = threadIdx.x & 31;
  int row0 = wave << 4;
  if (row0 >= n) return;                 // wave-uniform

  int m     = lane & 15;                 // A row within tile / C column
  int khalf = (lane >> 4) << 1;          // K sub-offset: 0 or 2
  int row   = row0 + m; if (row >= n) row = n - 1;
  const float* arow = in + (size_t)row * 16;

  v2f a[CHUNKS], b[CHUNKS];
#pragma unroll
  for (int c = 0; c < CHUNKS; ++c) {
    a[c] = *(const v2f*)(arow + c * 4 + khalf);      // 8B-aligned b64 load
    b[c] = *(const v2f*)(Bp + c * 64 + lane * 2);    // 8B-aligned b64 load
  }

  v8f acc = {};
#pragma unroll
  for (int c = 0; c < CHUNKS; ++c)
    acc = __builtin_amdgcn_wmma_f32_16x16x4_f32(
        /*neg_a=*/false, a[c], /*neg_b=*/false, b[c],
        /*c_mod=*/(short)0, acc, /*reuse_a=*/false, /*reuse_b=*/false);

  int mbase = (lane >> 4) << 3;          // C rows: r / r+8 split by lane half
  float* hp = h + (size_t)(row0 + mbase) * 16 + m;
  if (row0 + 16 <= n) {                  // wave-uniform full-tile fast path
#pragma unroll
    for (int r = 0; r < 8; ++r) hp[(size_t)r * 16] = acc[r];
  } else {                               // rare partial tile
#pragma unroll
    for (int r = 0; r < 8; ++r)
      if (row0 + mbase + r < n) hp[(size_t)r * 16] = acc[r];
  }
}

// ------------------------------------------------- edge scatter (L2 atomics)
// One thread per (edge, feature-lane): 16 lanes of an edge read 64 contiguous
// bytes of h[src] and issue global_atomic_add_f32 into agg[dst].
__global__ void scatter_kernel(const long long* __restrict__ src,
                               const long long* __restrict__ dst,
                               const float* __restrict__ dinv,
                               const float* __restrict__ h,
                               float* __restrict__ agg,
                               int fout, long long nedges) {
  long long t = (long long)blockIdx.x * blockDim.x + threadIdx.x;
  long long e = t >> 4;
  int f = (int)(t & 15);
  if (e >= nedges || f >= fout) return;
  int s = (int)src[e], d = (int)dst[e];
  float nrm = dinv[s] * dinv[d];
  atomicAdd(agg + (size_t)d * 16 + f, nrm * h[(size_t)s * 16 + f]);
}

// ------------------------------------------------- self-loop + bias + relu
__global__ void finalize_kernel(const float* __restrict__ h,
                                const float* __restrict__ agg,
                                const float* __restrict__ dinv,
                                const float* __restrict__ b,
                                float* __restrict__ out,
                                int fout, int dorelu, int n) {
  long long t = (long long)blockIdx.x * blockDim.x + threadIdx.x;
  int i = (int)(t >> 4);
  int f = (int)(t & 15);
  if (i >= n) return;
  float v = 0.f;
  if (f < fout) {
    float di = dinv[i];
    v = agg[(size_t)i * 16 + f] + di * di * h[(size_t)i * 16 + f] + b[f];
    if (dorelu) v = fmaxf(v, 0.f);
  }
  out[(size_t)i * 16 + f] = v;           // pad columns written as exact zeros
}

// ---------------------------------------------------------------- driver

static void run_layer(const float* in16, int fout, const float* Bp,
                      const float* bias, const long long* src,
                      const long long* dst, const float* dinv, float* H,
                      float* AGG, float* out, int n, long long E, int dorelu,
                      hipStream_t stream) {
  int waves = (n + 15) / 16;
  int mmblk = (waves + 7) / 8;                          // 8 waves / 256-thr block
  matmul_wmma_kernel<3><<<mmblk, 256, 0, stream>>>(in16, Bp, H, n);

  long long aggN = (long long)n * 16;
  zero_kernel<<<(int)((aggN + 255) / 256), 256, 0, stream>>>(AGG, aggN);

  long long st = E * 16;
  scatter_kernel<<<(int)((st + 255) / 256), 256, 0, stream>>>(src, dst, dinv, H,
                                                              AGG, fout, E);

  long long fn = (long long)n * 16;
  finalize_kernel<<<(int)((fn + 255) / 256), 256, 0, stream>>>(H, AGG, dinv,
                                                               bias, out, fout,
                                                               dorelu, n);
}

extern "C" void kernel_launch(void* const* d_in, const int* in_sizes, int n_in,
                              void* d_out, int out_size, void* d_ws,
                              size_t ws_size, hipStream_t stream) {
  const float*     x  = (const float*)d_in[0];
  const long long* ei = (const long long*)d_in[1];      // int64 [2, E]
  const float*     W1 = (const float*)d_in[2];
  const float*     b1 = (const float*)d_in[3];
  const float*     W2 = (const float*)d_in[4];
  const float*     b2 = (const float*)d_in[5];
  const float*     W3 = (const float*)d_in[6];
  const float*     b3 = (const float*)d_in[7];

  int       n = in_sizes[0] / 11;                       // 200000
  long long E = (long long)in_sizes[1] / 2;             // 6400000
  const long long* src = ei;
  const long long* dst = ei + E;

  // workspace layout (floats)
  float* ws   = (float*)d_ws;
  size_t off  = 0;
  float* dinv = ws + off; off += ((size_t)n + 255) & ~(size_t)255;
  float* B1p  = ws + off; off += 256;                   // 3 chunks * 64 packed
  float* B2p  = ws + off; off += 256;
  float* B3p  = ws + off; off += 256;
  float* X16  = ws + off; off += (size_t)n * 16;
  float* H    = ws + off; off += (size_t)n * 16;
  float* AGG  = ws + off; off += (size_t)n * 16;
  float* Y    = ws + off; off += (size_t)n * 16;

  // degrees + dinv (shared by all layers; depends only on dst)
  zero_kernel<<<(n + 255) / 256, 256, 0, stream>>>(dinv, n);
  degree_kernel<<<(int)((E + 255) / 256), 256, 0, stream>>>(dst, dinv, E);
  dinv_kernel<<<(n + 255) / 256, 256, 0, stream>>>(dinv, n);

  // pad x to stride 16; pack weights into WMMA B lane order
  long long px = (long long)n * 16;
  pad_x_kernel<<<(int)((px + 255) / 256), 256, 0, stream>>>(x, X16, n);
  pack_w_kernel<<<1, 192, 0, stream>>>(W1, B1p, 11, 11, 3);
  pack_w_kernel<<<1, 192, 0, stream>>>(W2, B2p, 11, 11, 3);
  pack_w_kernel<<<1, 192, 0, stream>>>(W3, B3p, 11, 16, 3);

  // layer 1: X16 -> Y, relu
  run_layer(X16, 11, B1p, b1, src, dst, dinv, H, AGG, Y, n, E, 1, stream);
  // layer 2: Y -> Y (matmul consumes Y before finalize overwrites it), relu
  run_layer(Y, 11, B2p, b2, src, dst, dinv, H, AGG, Y, n, E, 1, stream);
  // layer 3: Y -> d_out (N x 16), no relu
  run_layer(Y, 16, B3p, b3, src, dst, dinv, H, AGG, (float*)d_out, n, E, 0,
            stream);
}